// VAEBM_14800457302261
// MI455X (gfx1250) — compile-verified
//
#include <hip/hip_runtime.h>
#include <cstdint>

typedef float v2f __attribute__((ext_vector_type(2)));
typedef float v8f __attribute__((ext_vector_type(8)));

#define HID   100
#define STEPS 100

__device__ __forceinline__ float sigf(float a) {
    return 1.0f / (1.0f + __expf(-a));
}

// ---------------------------------------------------------------------------
// Encoder: h1 = swish(x@We1+be1); h2 = swish(h1@We2+be2) via WMMA f32 16x16x4;
// mu_zx = h2@Wemu+bemu, log_t = h2@Wet+bet.
// Block: 128 threads = 4 waves; 64 points per block; each wave owns a 16-row
// tile and sweeps 7 column tiles of 16 (100 -> padded 112).
// ---------------------------------------------------------------------------
__global__ void encoder_kernel(const float* __restrict__ x,
                               const float* __restrict__ We1, const float* __restrict__ be1,
                               const float* __restrict__ We2, const float* __restrict__ be2,
                               const float* __restrict__ Wemu, const float* __restrict__ bemu,
                               const float* __restrict__ Wet,  const float* __restrict__ bet,
                               float* __restrict__ mu_zx, float* __restrict__ log_t,
                               int n)
{
    __shared__ float h1s[64][HID];
    __shared__ float h2s[64][HID];
    const int tid  = threadIdx.x;
    const int base = blockIdx.x * 64;

    // h1 = swish(x @ We1 + be1)   (K=3: scalar is optimal)
    for (int idx = tid; idx < 64 * HID; idx += 128) {
        int p = idx / HID, h = idx % HID;
        int gp = base + p;
        float v = 0.0f;
        if (gp < n) {
            float a = x[gp*3+0]*We1[h] + x[gp*3+1]*We1[HID+h]
                    + x[gp*3+2]*We1[2*HID+h] + be1[h];
            v = a * sigf(a);
        }
        h1s[p][h] = v;
    }
    __syncthreads();

    const int wave = tid >> 5, lane = tid & 31;
    const int half = lane >> 4, pl = lane & 15;
    const int rb   = wave * 16;
    const int arow = rb + pl;

    for (int ct = 0; ct < 7; ++ct) {
        const int cb   = ct * 16;
        const int col  = cb + pl;
        const int colc = (col < HID) ? col : (HID - 1);
        const float cmask = (col < HID) ? 1.0f : 0.0f;
        v8f acc = {0.f,0.f,0.f,0.f,0.f,0.f,0.f,0.f};
        for (int t = 0; t < 25; ++t) {
            const int kA = 4*t + 2*half;          // A: K = 2*half + vgpr
            v2f A, B;
            A.x = h1s[arow][kA];
            A.y = h1s[arow][kA + 1];
            B.x = We2[(4*t + half)     * HID + colc] * cmask;  // B: K = 2*vgpr + half
            B.y = We2[(4*t + 2 + half) * HID + colc] * cmask;
            acc = __builtin_amdgcn_wmma_f32_16x16x4_f32(false, A, false, B,
                                                        (short)0, acc, false, false);
        }
        if (col < HID) {
            float bb = be2[col];
            #pragma unroll
            for (int v = 0; v < 8; ++v) {
                int prow = rb + v + 8*half;        // D: M = vgpr + 8*half
                float a = acc[v] + bb;
                h2s[prow][col] = a * sigf(a);
            }
        }
    }
    __syncthreads();

    if (tid < 64) {
        int gp = base + tid;
        float m0 = 0.f, m1 = 0.f, lt = 0.f;
        for (int h = 0; h < HID; ++h) {
            float hv = h2s[tid][h];
            m0 += hv * Wemu[h*2+0];
            m1 += hv * Wemu[h*2+1];
            lt += hv * Wet[h];
        }
        if (gp < n) {
            mu_zx[gp*2+0] = m0 + bemu[0];
            mu_zx[gp*2+1] = m1 + bemu[1];
            log_t[gp]     = lt + bet[0];
        }
    }
}

// ---------------------------------------------------------------------------
// Riemannian Brownian-motion sampler. 100 Euler-Maruyama steps fused in-kernel.
// Per wave: 16 points. Per step: dsw/ddsw activations go into WMMA A-fragments;
// two f32 16x16x4 WMMA chains (K=100) against constant coefficient tables give
// all 20 metric statistics {M, r, T, q}; 16 lanes finish G^-1 / Christoffel /
// drift / z-update.
// ---------------------------------------------------------------------------
__global__ void sampler_kernel(const float* __restrict__ Wd1, const float* __restrict__ bd1,
                               const float* __restrict__ Wdmu, const float* __restrict__ Wds,
                               const float* __restrict__ mu_init, const float* __restrict__ logt_init,
                               const float* __restrict__ prior_mu, const float* __restrict__ prior_t,
                               const float* __restrict__ eps,
                               float* __restrict__ z_out,
                               int n, int use_prior)
{
    __shared__ float wpk[HID][4];       // w0, w1, bd1, pad
    __shared__ float B1t[HID][16];      // dsw-stat coeffs: M(6), r(2), zero pad
    __shared__ float B2t[HID][16];      // ddsw-stat coeffs: T(9), q(3), zero pad
    __shared__ float zbuf[64][2];
    __shared__ float sqdt[64];
    __shared__ float st1[4][16][16];
    __shared__ float st2[4][16][16];

    const int tid  = threadIdx.x;
    const int base = blockIdx.x * 64;

    if (tid < HID) {
        const int h = tid;
        float w0 = Wd1[h], w1 = Wd1[HID + h];
        wpk[h][0] = w0; wpk[h][1] = w1; wpk[h][2] = bd1[h]; wpk[h][3] = 0.f;
        float u0 = Wdmu[h*3+0], u1 = Wdmu[h*3+1], u2 = Wdmu[h*3+2], vv = Wds[h];
        float w00 = w0*w0, w01 = w0*w1, w11 = w1*w1;
        B1t[h][0] = u0*w0; B1t[h][1] = u0*w1;
        B1t[h][2] = u1*w0; B1t[h][3] = u1*w1;
        B1t[h][4] = u2*w0; B1t[h][5] = u2*w1;
        B1t[h][6] = vv*w0; B1t[h][7] = vv*w1;
        #pragma unroll
        for (int c = 8; c < 16; ++c) B1t[h][c] = 0.f;
        B2t[h][0] = u0*w00; B2t[h][1] = u0*w01; B2t[h][2]  = u0*w11;
        B2t[h][3] = u1*w00; B2t[h][4] = u1*w01; B2t[h][5]  = u1*w11;
        B2t[h][6] = u2*w00; B2t[h][7] = u2*w01; B2t[h][8]  = u2*w11;
        B2t[h][9] = vv*w00; B2t[h][10] = vv*w01; B2t[h][11] = vv*w11;
        #pragma unroll
        for (int c = 12; c < 16; ++c) B2t[h][c] = 0.f;
    }
    if (tid < 64) {
        int gp  = base + tid;
        int gpc = (gp < n) ? gp : (n - 1);
        float z0, z1, lt;
        if (use_prior) { z0 = prior_mu[0]; z1 = prior_mu[1]; lt = prior_t[0]; }
        else { z0 = mu_init[gpc*2+0]; z1 = mu_init[gpc*2+1]; lt = logt_init[gpc]; }
        float tt = __expf(2.0f * lt);
        bool ok = (gp < n);
        zbuf[tid][0] = ok ? z0 : 0.f;
        zbuf[tid][1] = ok ? z1 : 0.f;
        sqdt[tid]    = ok ? sqrtf(tt / (float)STEPS) : 0.f;
    }
    __syncthreads();

    const int wave = tid >> 5, lane = tid & 31;
    const int half = lane >> 4, pl = lane & 15;
    const int plocal = wave * 16 + pl;
    const int gp  = base + plocal;
    const int gpc = (gp < n) ? gp : (n - 1);

    for (int step = 0; step < STEPS; ++step) {
        float z0 = zbuf[plocal][0];
        float z1 = zbuf[plocal][1];

        v8f acc1 = {0.f,0.f,0.f,0.f,0.f,0.f,0.f,0.f};
        v8f acc2 = {0.f,0.f,0.f,0.f,0.f,0.f,0.f,0.f};
        for (int t = 0; t < 25; ++t) {
            const int hA = 4*t + 2*half;           // A lanes: K = 2*half + vgpr
            float w0a = wpk[hA][0],   w1a = wpk[hA][1],   ba = wpk[hA][2];
            float w0b = wpk[hA+1][0], w1b = wpk[hA+1][1], bb = wpk[hA+1][2];
            float a0 = w0a*z0 + w1a*z1 + ba;
            float a1 = w0b*z0 + w1b*z1 + bb;
            float s0 = sigf(a0), s1 = sigf(a1);
            float sm0 = s0*(1.0f - s0), sm1 = s1*(1.0f - s1);
            v2f A1, A2, Bf1, Bf2;
            A1.x = s0 + a0*sm0;                         // swish'(a)
            A1.y = s1 + a1*sm1;
            A2.x = sm0*(2.0f + a0*(1.0f - 2.0f*s0));    // swish''(a)
            A2.y = sm1*(2.0f + a1*(1.0f - 2.0f*s1));
            const int kB0 = 4*t + half, kB1 = 4*t + 2 + half;  // B: K = 2*vgpr + half
            Bf1.x = B1t[kB0][pl]; Bf1.y = B1t[kB1][pl];
            Bf2.x = B2t[kB0][pl]; Bf2.y = B2t[kB1][pl];
            acc1 = __builtin_amdgcn_wmma_f32_16x16x4_f32(false, A1, false, Bf1,
                                                         (short)0, acc1, false, false);
            acc2 = __builtin_amdgcn_wmma_f32_16x16x4_f32(false, A2, false, Bf2,
                                                         (short)0, acc2, false, false);
        }
        #pragma unroll
        for (int v = 0; v < 8; ++v) {                  // D: point = vgpr + 8*half, stat = pl
            int prow = v + 8*half;
            st1[wave][prow][pl] = acc1[v];
            st2[wave][prow][pl] = acc2[v];
        }
        __syncthreads();

        if (half == 0) {
            float Ms[3][2], rs[2], Ts[3][3], qs[3];
            Ms[0][0] = st1[wave][pl][0]; Ms[0][1] = st1[wave][pl][1];
            Ms[1][0] = st1[wave][pl][2]; Ms[1][1] = st1[wave][pl][3];
            Ms[2][0] = st1[wave][pl][4]; Ms[2][1] = st1[wave][pl][5];
            rs[0]    = st1[wave][pl][6]; rs[1]    = st1[wave][pl][7];
            Ts[0][0] = st2[wave][pl][0]; Ts[0][1] = st2[wave][pl][1]; Ts[0][2] = st2[wave][pl][2];
            Ts[1][0] = st2[wave][pl][3]; Ts[1][1] = st2[wave][pl][4]; Ts[1][2] = st2[wave][pl][5];
            Ts[2][0] = st2[wave][pl][6]; Ts[2][1] = st2[wave][pl][7]; Ts[2][2] = st2[wave][pl][8];
            qs[0]    = st2[wave][pl][9]; qs[1]    = st2[wave][pl][10]; qs[2]   = st2[wave][pl][11];

            float g00 = Ms[0][0]*Ms[0][0] + Ms[1][0]*Ms[1][0] + Ms[2][0]*Ms[2][0] + 3.0f*rs[0]*rs[0];
            float g01 = Ms[0][0]*Ms[0][1] + Ms[1][0]*Ms[1][1] + Ms[2][0]*Ms[2][1] + 3.0f*rs[0]*rs[1];
            float g11 = Ms[0][1]*Ms[0][1] + Ms[1][1]*Ms[1][1] + Ms[2][1]*Ms[2][1] + 3.0f*rs[1]*rs[1];
            float inv = 1.0f / (g00*g11 - g01*g01);
            float gi[2][2] = {{ g11*inv, -g01*inv}, {-g01*inv, g00*inv}};

            float DG[2][2][2];
            #pragma unroll
            for (int k = 0; k < 2; ++k)
              #pragma unroll
              for (int m = 0; m < 2; ++m)
                #pragma unroll
                for (int l = 0; l < 2; ++l) {
                    float acc = 3.0f*(qs[k+l]*rs[m] + rs[k]*qs[m+l]);
                    #pragma unroll
                    for (int i = 0; i < 3; ++i)
                        acc += Ts[i][k+l]*Ms[i][m] + Ms[i][k]*Ts[i][m+l];
                    DG[k][m][l] = acc;
                }
            float Ch[2][2][2];
            #pragma unroll
            for (int i2 = 0; i2 < 2; ++i2)
              #pragma unroll
              for (int k = 0; k < 2; ++k)
                #pragma unroll
                for (int l = 0; l < 2; ++l) {
                    float acc = 0.f;
                    #pragma unroll
                    for (int m = 0; m < 2; ++m)
                        acc += gi[i2][m]*(DG[k][m][l] + DG[l][m][k] - DG[k][l][m]);
                    Ch[i2][k][l] = 0.5f * acc;
                }
            float dr0 = 0.5f*(gi[0][0]*Ch[0][0][0] + gi[0][1]*Ch[0][0][1]
                            + gi[1][0]*Ch[0][1][0] + gi[1][1]*Ch[0][1][1]);
            float dr1 = 0.5f*(gi[0][0]*Ch[1][0][0] + gi[0][1]*Ch[1][0][1]
                            + gi[1][0]*Ch[1][1][0] + gi[1][1]*Ch[1][1][1]);

            float e0 = eps[((size_t)step*(size_t)n + (size_t)gpc)*2 + 0];
            float e1 = eps[((size_t)step*(size_t)n + (size_t)gpc)*2 + 1];
            float sq = sqdt[plocal];
            float dW0 = sq*e0, dW1 = sq*e1;
            zbuf[plocal][0] = z0 + dr0 + gi[0][0]*dW0 + gi[0][1]*dW1;
            zbuf[plocal][1] = z1 + dr1 + gi[1][0]*dW0 + gi[1][1]*dW1;
        }
        __syncthreads();
    }

    if (tid < 64) {
        int g = base + tid;
        if (g < n) {
            z_out[g*2+0] = zbuf[tid][0];
            z_out[g*2+1] = zbuf[tid][1];
        }
    }
}

// ---------------------------------------------------------------------------
// Epilogue: decoder outputs, G(mu_zx), G(mu_z), prior broadcasts, dt / dW.
// ---------------------------------------------------------------------------
__device__ void Gcalc(float z0, float z1,
                      const float* __restrict__ Wd1, const float* __restrict__ bd1,
                      const float* __restrict__ Wdmu, const float* __restrict__ Wds,
                      float& g00, float& g01, float& g11)
{
    float M00=0,M01=0,M10=0,M11=0,M20=0,M21=0, r0=0, r1=0;
    for (int h = 0; h < HID; ++h) {
        float w0 = Wd1[h], w1 = Wd1[HID + h];
        float a = w0*z0 + w1*z1 + bd1[h];
        float s = sigf(a);
        float dsw = s + a*s*(1.0f - s);
        float p0 = dsw*Wdmu[h*3+0], p1 = dsw*Wdmu[h*3+1], p2 = dsw*Wdmu[h*3+2];
        M00 += p0*w0; M01 += p0*w1;
        M10 += p1*w0; M11 += p1*w1;
        M20 += p2*w0; M21 += p2*w1;
        float pv = dsw*Wds[h]; r0 += pv*w0; r1 += pv*w1;
    }
    g00 = M00*M00 + M10*M10 + M20*M20 + 3.0f*r0*r0;
    g01 = M00*M01 + M10*M11 + M20*M21 + 3.0f*r0*r1;
    g11 = M01*M01 + M11*M11 + M21*M21 + 3.0f*r1*r1;
}

__global__ void epilogue_kernel(const float* __restrict__ Wd1, const float* __restrict__ bd1,
                                const float* __restrict__ Wdmu, const float* __restrict__ bdmu,
                                const float* __restrict__ Wds,  const float* __restrict__ bds,
                                const float* __restrict__ bpm,  const float* __restrict__ bpt,
                                const float* __restrict__ eps_post,
                                const float* __restrict__ eps_prior,
                                float* __restrict__ out, int n)
{
    int gp = blockIdx.x * blockDim.x + threadIdx.x;
    if (gp >= n) return;
    size_t N = (size_t)n;
    float* z      = out;
    float* mu_xz  = out + 2*N;
    float* ls     = out + 5*N;
    const float* mu_zx = out + 8*N;
    const float* logt  = out + 10*N;
    float* mu_z   = out + 11*N;
    float* logtz  = out + 13*N;
    float* dW     = out + 14*N;
    float* dt_o   = out + 16*N;
    float* dWp    = out + 19*N;
    float* dtp_o  = out + 21*N;
    float* Gzx    = out + 22*N;
    float* Gz     = out + 26*N;

    // decoder at final posterior z
    float z0 = z[gp*2+0], z1 = z[gp*2+1];
    float m0=0,m1=0,m2=0,l=0;
    for (int h = 0; h < HID; ++h) {
        float a = Wd1[h]*z0 + Wd1[HID+h]*z1 + bd1[h];
        float hs = a * sigf(a);
        m0 += hs*Wdmu[h*3+0]; m1 += hs*Wdmu[h*3+1]; m2 += hs*Wdmu[h*3+2];
        l  += hs*Wds[h];
    }
    mu_xz[gp*3+0] = m0 + bdmu[0];
    mu_xz[gp*3+1] = m1 + bdmu[1];
    mu_xz[gp*3+2] = m2 + bdmu[2];
    float lsv = l + bds[0];
    ls[gp*3+0] = lsv; ls[gp*3+1] = lsv; ls[gp*3+2] = lsv;

    // G at mu_zx
    float g00,g01,g11;
    Gcalc(mu_zx[gp*2+0], mu_zx[gp*2+1], Wd1, bd1, Wdmu, Wds, g00, g01, g11);
    Gzx[gp*4+0]=g00; Gzx[gp*4+1]=g01; Gzx[gp*4+2]=g01; Gzx[gp*4+3]=g11;

    // prior params + G at mu_z
    float p0 = bpm[0], p1 = bpm[1], pt = bpt[0];
    mu_z[gp*2+0] = p0; mu_z[gp*2+1] = p1; logtz[gp] = pt;
    Gcalc(p0, p1, Wd1, bd1, Wdmu, Wds, g00, g01, g11);
    Gz[gp*4+0]=g00; Gz[gp*4+1]=g01; Gz[gp*4+2]=g01; Gz[gp*4+3]=g11;

    // dt and last-step dW (posterior and prior)
    float tt = __expf(2.0f * logt[gp]);
    float dt = tt / (float)STEPS, sq = sqrtf(dt);
    dt_o[gp] = dt;
    size_t le = ((size_t)(STEPS-1)*N + (size_t)gp)*2;
    dW[gp*2+0] = sq * eps_post[le + 0];
    dW[gp*2+1] = sq * eps_post[le + 1];
    float ttp = __expf(2.0f * pt);
    float dtp = ttp / (float)STEPS, sqp = sqrtf(dtp);
    dtp_o[gp] = dtp;
    dWp[gp*2+0] = sqp * eps_prior[le + 0];
    dWp[gp*2+1] = sqp * eps_prior[le + 1];
}

// ---------------------------------------------------------------------------
extern "C" void kernel_launch(void* const* d_in, const int* in_sizes, int n_in,
                              void* d_out, int out_size, void* d_ws, size_t ws_size,
                              hipStream_t stream) {
    const float* x    = (const float*)d_in[0];
    const float* We1  = (const float*)d_in[1];
    const float* be1  = (const float*)d_in[2];
    const float* We2  = (const float*)d_in[3];
    const float* be2  = (const float*)d_in[4];
    const float* Wemu = (const float*)d_in[5];
    const float* bemu = (const float*)d_in[6];
    const float* Wet  = (const float*)d_in[7];
    const float* bet  = (const float*)d_in[8];
    const float* Wd1  = (const float*)d_in[9];
    const float* bd1  = (const float*)d_in[10];
    const float* Wdmu = (const float*)d_in[11];
    const float* bdmu = (const float*)d_in[12];
    const float* Wds  = (const float*)d_in[13];
    const float* bds  = (const float*)d_in[14];
    const float* bpm  = (const float*)d_in[15];
    const float* bpt  = (const float*)d_in[16];
    const float* eps_post  = (const float*)d_in[17];
    const float* eps_prior = (const float*)d_in[18];

    const int n = in_sizes[0] / 3;
    const size_t N = (size_t)n;
    float* out = (float*)d_out;

    float* o_z     = out;
    float* o_muzx  = out + 8*N;
    float* o_logt  = out + 10*N;
    float* o_zpr   = out + 17*N;

    const int blocks64 = (n + 63) / 64;

    encoder_kernel<<<blocks64, 128, 0, stream>>>(
        x, We1, be1, We2, be2, Wemu, bemu, Wet, bet, o_muzx, o_logt, n);

    sampler_kernel<<<blocks64, 128, 0, stream>>>(
        Wd1, bd1, Wdmu, Wds, o_muzx, o_logt, bpm, bpt, eps_post, o_z, n, 0);

    sampler_kernel<<<blocks64, 128, 0, stream>>>(
        Wd1, bd1, Wdmu, Wds, o_muzx, o_logt, bpm, bpt, eps_prior, o_zpr, n, 1);

    epilogue_kernel<<<(n + 255) / 256, 256, 0, stream>>>(
        Wd1, bd1, Wdmu, bdmu, Wds, bds, bpm, bpt, eps_post, eps_prior, out, n);
}